// InfoGuidanceModule_76184129896983
// MI455X (gfx1250) — compile-verified
//
#include <hip/hip_runtime.h>
#include <math.h>

typedef __attribute__((ext_vector_type(2))) float v2f;
typedef __attribute__((ext_vector_type(8))) float v8f;

#define EPS_MI 1e-9f
#define EPS_BN 1e-5f

// ---------------------------------------------------------------------------
// WMMA GEMM + bias + BN + ReLU.
//   out[b][o][n] = relu( ((sum_c W[o][c]*X[b][c][n] + bias[o]) - rm[o]) *
//                        rsqrt(rv[o]+eps) * g[o] + beta[o] )
// M = 256 output channels, N = 256 spatial, K = 256 or 512 input channels.
// One wave per 16x16 output tile, V_WMMA_F32_16X16X4_F32 over K in steps of 4.
// grid = (16 Mtiles, 4 Ntile-groups, B), block = 128 (4 waves).
// ---------------------------------------------------------------------------
__global__ __launch_bounds__(128)
void gemm_bn_relu_kernel(const float* __restrict__ X,   // [B][K][256]
                         const float* __restrict__ W,   // [256][K]
                         const float* __restrict__ bias,
                         const float* __restrict__ rm,
                         const float* __restrict__ rv,
                         const float* __restrict__ gam,
                         const float* __restrict__ bet,
                         float* __restrict__ out,       // [B][256][256]
                         int K)
{
    const int N = 256;
    const int wave = threadIdx.x >> 5;
    const int lane = threadIdx.x & 31;
    const int mt = blockIdx.x;                  // 0..15
    const int nt = (blockIdx.y << 2) | wave;    // 0..15
    const int b  = blockIdx.z;

    const float* Xb = X + (size_t)b * K * N;

    const int l16   = lane & 15;
    const int khalf = (lane >> 4) << 1;         // 0 or 2

    // A fragment: lane holds row M = mt*16 + l16, regs = K = khalf, khalf+1
    const float* Arow = W + (size_t)(mt * 16 + l16) * K + khalf;
    // B fragment: lane holds col N = nt*16 + l16, regs = K = khalf, khalf+1
    const float* Bcol = Xb + (size_t)khalf * N + (nt * 16 + l16);

    v8f acc = {};
    for (int k0 = 0; k0 < K; k0 += 4) {
        v2f a = *reinterpret_cast<const v2f*>(Arow);   // 8B-aligned (khalf even)
        v2f bb;
        bb.x = Bcol[0];
        bb.y = Bcol[N];
        acc = __builtin_amdgcn_wmma_f32_16x16x4_f32(
            /*neg_a=*/false, a, /*neg_b=*/false, bb,
            /*c_mod=*/(short)0, acc, /*reuse_a=*/false, /*reuse_b=*/false);
        Arow += 4;
        Bcol += 4 * N;
    }

    // C/D layout: reg r, lanes 0-15 -> M = r, lanes 16-31 -> M = r + 8; N = l16
    const int mbase = mt * 16 + ((lane >> 4) << 3);
    const int ncol  = nt * 16 + l16;
    float* Ob = out + ((size_t)b * 256 + mbase) * N + ncol;
#pragma unroll
    for (int r = 0; r < 8; ++r) {
        const int o = mbase + r;
        const float inv = rsqrtf(rv[o] + EPS_BN);
        float y = (acc[r] + bias[o] - rm[o]) * inv * gam[o] + bet[o];
        Ob[(size_t)r * N] = fmaxf(y, 0.0f);
    }
}

// ---------------------------------------------------------------------------
// Softmax (over N=256) + entropy per (src, b, c) row; one wave per row.
// Stores probabilities (for MI) and entropy concat [B][512] (vis | text).
// grid = 512 blocks * 256 threads (8 waves) covers 2*8*256 = 4096 rows.
// ---------------------------------------------------------------------------
__global__ __launch_bounds__(256)
void softmax_entropy_kernel(const float* __restrict__ vis,
                            const float* __restrict__ text,
                            float* __restrict__ vn,
                            float* __restrict__ tn,
                            float* __restrict__ ent)   // [B][512]
{
    const int wave = threadIdx.x >> 5;
    const int lane = threadIdx.x & 31;
    const int row  = blockIdx.x * 8 + wave;     // 0..4095
    const int src  = row >> 11;                 // 0 = vis, 1 = text
    const int bc   = row & 2047;

    const float* in   = (src ? text : vis) + (size_t)bc * 256;
    float*       pout = (src ? tn : vn) + (size_t)bc * 256;

    float x[8];
    float mx = -INFINITY;
#pragma unroll
    for (int i = 0; i < 8; ++i) { x[i] = in[lane + 32 * i]; mx = fmaxf(mx, x[i]); }
    for (int off = 16; off; off >>= 1) mx = fmaxf(mx, __shfl_xor(mx, off, 32));

    float s = 0.0f;
#pragma unroll
    for (int i = 0; i < 8; ++i) { x[i] = __expf(x[i] - mx); s += x[i]; }
    for (int off = 16; off; off >>= 1) s += __shfl_xor(s, off, 32);

    const float inv = 1.0f / s;
    float h = 0.0f;
#pragma unroll
    for (int i = 0; i < 8; ++i) {
        const float p = x[i] * inv;
        pout[lane + 32 * i] = p;
        h -= p * __logf(p + EPS_MI);
    }
    for (int off = 16; off; off >>= 1) h += __shfl_xor(h, off, 32);

    if (lane == 0) {
        const int b = bc >> 8, c = bc & 255;
        ent[b * 512 + src * 256 + c] = h;
    }
}

// ---------------------------------------------------------------------------
// Mutual information per (b,c): block of 256 threads, thread j owns column j.
//   mi = sum_ij (vn_i*tn_j/S) * log(vn_i*tn_j/(S*denom) + eps)
//   S = p_v*p_t, denom = S + eps
// ---------------------------------------------------------------------------
__global__ __launch_bounds__(256)
void mi_kernel(const float* __restrict__ vn,
               const float* __restrict__ tn,
               float* __restrict__ mi)          // [B*256]
{
    __shared__ float sv[256];
    __shared__ float st[256];
    __shared__ float red[256];
    const int bc = blockIdx.x;
    const int t  = threadIdx.x;

    sv[t] = vn[(size_t)bc * 256 + t];
    st[t] = tn[(size_t)bc * 256 + t];
    __syncthreads();

    // p_v, p_t via broadcast reads (same-address LDS reads broadcast)
    float pv = 0.0f, pt = 0.0f;
    for (int i = 0; i < 256; ++i) { pv += sv[i]; pt += st[i]; }

    const float S     = pv * pt;
    const float invS  = 1.0f / S;
    const float c1    = invS / (S + EPS_MI);    // joint/denom = vn*tn*c1
    const float tj    = st[t];

    float acc = 0.0f;
    for (int i = 0; i < 256; ++i) {
        const float p = sv[i] * tj;
        acc += p * __logf(p * c1 + EPS_MI);
    }
    acc *= invS;

    red[t] = acc;
    __syncthreads();
    for (int sdx = 128; sdx; sdx >>= 1) {
        if (t < sdx) red[t] += red[t + sdx];
        __syncthreads();
    }
    if (t == 0) mi[bc] = red[0];
}

// ---------------------------------------------------------------------------
// Generic 2-layer MLP + sigmoid:  out = sigmoid(W2 @ relu(W1 @ in + b1) + b2)
// One block (256 threads) per batch; thread t owns output neuron t.
// ---------------------------------------------------------------------------
__global__ __launch_bounds__(256)
void mlp_sigmoid_kernel(const float* __restrict__ in, int in_dim,
                        const float* __restrict__ W1, const float* __restrict__ b1,
                        const float* __restrict__ W2, const float* __restrict__ b2,
                        float* __restrict__ out)    // [B][256]
{
    __shared__ float s_in[512];
    __shared__ float s_h[256];
    const int b = blockIdx.x;
    const int t = threadIdx.x;

    const float* inb = in + (size_t)b * in_dim;
    for (int i = t; i < in_dim; i += 256) s_in[i] = inb[i];
    __syncthreads();

    const float* w1 = W1 + (size_t)t * in_dim;
    float a1 = b1[t];
    for (int i = 0; i < in_dim; ++i) a1 = fmaf(w1[i], s_in[i], a1);
    s_h[t] = fmaxf(a1, 0.0f);
    __syncthreads();

    const float* w2 = W2 + (size_t)t * 256;
    float a2 = b2[t];
    for (int i = 0; i < 256; ++i) a2 = fmaf(w2[i], s_h[i], a2);
    out[(size_t)b * 256 + t] = 1.0f / (1.0f + __expf(-a2));
}

// ---------------------------------------------------------------------------
// Weighted concat: wx[b][c]      = vis*ew*mw
//                  wx[b][256+c]  = text*(1-ew)*mw
// ---------------------------------------------------------------------------
__global__ __launch_bounds__(256)
void weight_concat_kernel(const float* __restrict__ vis,
                          const float* __restrict__ text,
                          const float* __restrict__ ew,
                          const float* __restrict__ mw,
                          float* __restrict__ wx)   // [B][512][256]
{
    const int idx = blockIdx.x * 256 + threadIdx.x;  // over B*C*N = 524288
    const int n  = idx & 255;
    const int bc = idx >> 8;
    const int b  = bc >> 8;
    const int c  = bc & 255;
    const float e = ew[bc];
    const float m = mw[bc];
    wx[((size_t)b * 512 + c)       * 256 + n] = vis[idx]  * e * m;
    wx[((size_t)b * 512 + 256 + c) * 256 + n] = text[idx] * (1.0f - e) * m;
}

// ---------------------------------------------------------------------------
extern "C" void kernel_launch(void* const* d_in, const int* in_sizes, int n_in,
                              void* d_out, int out_size, void* d_ws, size_t ws_size,
                              hipStream_t stream) {
    const float* vis_feat  = (const float*)d_in[0];
    const float* text_feat = (const float*)d_in[1];
    const float* Wt  = (const float*)d_in[2];
    const float* bt  = (const float*)d_in[3];
    const float* g1  = (const float*)d_in[4];
    const float* b1  = (const float*)d_in[5];
    const float* m1  = (const float*)d_in[6];
    const float* v1  = (const float*)d_in[7];
    const float* We1 = (const float*)d_in[8];
    const float* be1 = (const float*)d_in[9];
    const float* We2 = (const float*)d_in[10];
    const float* be2 = (const float*)d_in[11];
    const float* Wm1 = (const float*)d_in[12];
    const float* bm1 = (const float*)d_in[13];
    const float* Wm2 = (const float*)d_in[14];
    const float* bm2 = (const float*)d_in[15];
    const float* Wf  = (const float*)d_in[16];
    const float* bf  = (const float*)d_in[17];
    const float* g2  = (const float*)d_in[18];
    const float* b2  = (const float*)d_in[19];
    const float* m2  = (const float*)d_in[20];
    const float* v2  = (const float*)d_in[21];

    float* ws = (float*)d_ws;
    float* vis_trans  = ws;                 // 524288
    float* text_trans = ws + 524288;        // 524288
    float* vn         = ws + 1048576;       // 524288
    float* tn         = ws + 1572864;       // 524288
    float* wx         = ws + 2097152;       // 1048576
    float* ent        = ws + 3145728;       // 4096  [B][512]
    float* ew         = ws + 3149824;       // 2048
    float* mi         = ws + 3151872;       // 2048
    float* mw         = ws + 3153920;       // 2048

    const dim3 gblk(128);
    const dim3 ggrid(16, 4, 8);

    // vis / text transform (shared weights)
    gemm_bn_relu_kernel<<<ggrid, gblk, 0, stream>>>(vis_feat,  Wt, bt, m1, v1, g1, b1, vis_trans, 256);
    gemm_bn_relu_kernel<<<ggrid, gblk, 0, stream>>>(text_feat, Wt, bt, m1, v1, g1, b1, text_trans, 256);

    // softmax probs + entropies
    softmax_entropy_kernel<<<512, 256, 0, stream>>>(vis_trans, text_trans, vn, tn, ent);

    // entropy weights
    mlp_sigmoid_kernel<<<8, 256, 0, stream>>>(ent, 512, We1, be1, We2, be2, ew);

    // mutual information + MI weights
    mi_kernel<<<2048, 256, 0, stream>>>(vn, tn, mi);
    mlp_sigmoid_kernel<<<8, 256, 0, stream>>>(mi, 256, Wm1, bm1, Wm2, bm2, mw);

    // weighted concat + fused conv (K = 512) -> d_out
    weight_concat_kernel<<<2048, 256, 0, stream>>>(vis_trans, text_trans, ew, mw, wx);
    gemm_bn_relu_kernel<<<ggrid, gblk, 0, stream>>>(wx, Wf, bf, m2, v2, g2, b2, (float*)d_out, 512);
}